// BinTokenizer_90812788507001
// MI455X (gfx1250) — compile-verified
//
#include <hip/hip_runtime.h>

// BinTokenizer for MI455X (gfx1250, wave32).
// Memory-bound streaming bucketize: float32 in -> int32 token out.
// CDNA5 paths: async global->LDS staging of thresholds (ASYNCcnt +
// s_wait_asynccnt), non-temporal B128 streaming loads/stores, LDS binary
// search with the top 3 levels hoisted into registers (v_cndmask tree).

#define TPB   256          // 8 wave32 waves per block
#define NBINS 256          // 257 threshold edges

typedef float v4f __attribute__((ext_vector_type(4)));
typedef int   v4i __attribute__((ext_vector_type(4)));

struct TopLevels {
  float t127;
  float t63, t191;
  float t31, t95, t159, t223;
  float t_last;              // t[256]
};

__device__ __forceinline__ int bin_search(const float* __restrict__ t,
                                          const TopLevels& tl, float x) {
  // clip(inputs, LOW+EPS, HIGH-EPS) as in the reference
  x = fminf(fmaxf(x, 1e-6f), 1.0f - 1e-6f);

  // searchsorted(side='right') over t[0..254]:
  //   "at least r+w elements <= x"  <=>  t[r+w-1] <= x
  // Levels 1-3 from registers (7 values, select tree), levels 4-8 from LDS.
  const bool c1 = (tl.t127 <= x);
  int r = c1 ? 128 : 0;

  const float v2 = c1 ? tl.t191 : tl.t63;
  const bool c2 = (v2 <= x);
  r += c2 ? 64 : 0;

  const float v3a = c1 ? tl.t159 : tl.t31;
  const float v3b = c1 ? tl.t223 : tl.t95;
  const float v3 = c2 ? v3b : v3a;
  r += (v3 <= x) ? 32 : 0;

  if (t[r + 15] <= x) r += 16;
  if (t[r + 7]  <= x) r += 8;
  if (t[r + 3]  <= x) r += 4;
  if (t[r + 1]  <= x) r += 2;
  if (t[r]      <= x) r += 1;

  if (tl.t_last <= x) r += 1;   // 257th edge (t[256])
  r -= 1;                        // searchsorted - 1
  r = r < 0 ? 0 : r;             // clip(tokens, 0, NBINS-1)
  r = r > (NBINS - 1) ? (NBINS - 1) : r;
  return r;
}

__global__ __launch_bounds__(TPB) void BinTokenizer_kernel(
    const float* __restrict__ in, const float* __restrict__ thr,
    int* __restrict__ out, long long n4, long long n) {
  __shared__ float t[NBINS + 1];
  const int tid = threadIdx.x;

  // ---- CDNA5 async global->LDS staging of the 257 thresholds ----
  {
    unsigned           lds_off = (unsigned)(unsigned long long)(void*)&t[tid];
    unsigned long long gaddr   = (unsigned long long)(thr + tid);
    asm volatile("global_load_async_to_lds_b32 %0, %1, off"
                 :: "v"(lds_off), "v"(gaddr) : "memory");
    if (tid == 0) {
      unsigned           lds_off2 = (unsigned)(unsigned long long)(void*)&t[NBINS];
      unsigned long long gaddr2   = (unsigned long long)(thr + NBINS);
      asm volatile("global_load_async_to_lds_b32 %0, %1, off"
                   :: "v"(lds_off2), "v"(gaddr2) : "memory");
    }
    asm volatile("s_wait_asynccnt 0x0" ::: "memory");
  }
  __syncthreads();

  // Hoist the top 3 binary-search levels (and the final edge) to registers.
  TopLevels tl;
  tl.t127 = t[127];
  tl.t63  = t[63];  tl.t191 = t[191];
  tl.t31  = t[31];  tl.t95  = t[95];
  tl.t159 = t[159]; tl.t223 = t[223];
  tl.t_last = t[NBINS];

  // ---- grid-stride streaming loop, 4 elements / lane / iteration ----
  long long i = (long long)blockIdx.x * TPB + tid;
  const long long stride = (long long)gridDim.x * TPB;
  const v4f* __restrict__ in4  = (const v4f*)in;
  v4i*       __restrict__ out4 = (v4i*)out;

  for (; i < n4; i += stride) {
    v4f v = __builtin_nontemporal_load(in4 + i);   // global_load_b128 NT
    v4i o;
    o.x = bin_search(t, tl, v.x);
    o.y = bin_search(t, tl, v.y);
    o.z = bin_search(t, tl, v.z);
    o.w = bin_search(t, tl, v.w);
    __builtin_nontemporal_store(o, out4 + i);      // global_store_b128 NT
  }

  // scalar tail (n not divisible by 4; not hit for this shape)
  for (long long j = n4 * 4 + (long long)blockIdx.x * TPB + tid; j < n;
       j += stride) {
    out[j] = bin_search(t, tl, in[j]);
  }
}

extern "C" void kernel_launch(void* const* d_in, const int* in_sizes, int n_in,
                              void* d_out, int out_size, void* d_ws,
                              size_t ws_size, hipStream_t stream) {
  const float* in  = (const float*)d_in[0];   // (64, 4096, 512) fp32
  const float* thr = (const float*)d_in[1];   // 257 fp32 edges
  int* out = (int*)d_out;                     // int32 tokens

  long long n  = (long long)in_sizes[0];
  long long n4 = n >> 2;

  long long blocks_needed = (n4 + TPB - 1) / TPB;
  if (blocks_needed < 1) blocks_needed = 1;
  int blocks = (int)(blocks_needed > 16384 ? 16384 : blocks_needed);

  BinTokenizer_kernel<<<blocks, TPB, 0, stream>>>(in, thr, out, n4, n);
}